// AttnDecoderRNN_26104811225140
// MI455X (gfx1250) — compile-verified
//
#include <hip/hip_runtime.h>
#include <math.h>

#define V_SZ 50257
#define H_SZ 4096
#define L_SZ 512
#define HH_SZ 2048

typedef __attribute__((ext_vector_type(2))) float v2f;
typedef __attribute__((ext_vector_type(8))) float v8f;

// ---------------------------------------------------------------------------
// WMMA GEMV: y[M] = W[M,K] @ v[K] + bias[M], optional ReLU.
// One wave per 16-row tile. K must be a multiple of 128 (true for all calls:
// 8192 / 4096 / 2048). Stages W tile + x chunk through LDS so global loads
// stay fully coalesced (float4 per lane, 512B bursts per row), then runs
// v_wmma_f32_16x16x4_f32 with A = W tile, B = x broadcast over all columns.
// ---------------------------------------------------------------------------
#define GEMV_WAVES 4
#define GEMV_BK 128
#define GEMV_ROWS 132              // padded LDS row stride (conflict-free)
#define GEMV_WAVE_LDS (16 * GEMV_ROWS + GEMV_BK)   // 2240 floats per wave

__global__ void gemv_wmma_kernel(const float* __restrict__ W,
                                 const float* __restrict__ v,
                                 const float* __restrict__ bias,
                                 float* __restrict__ y,
                                 int M, int K, int act) {
  __shared__ float lds[GEMV_WAVES * GEMV_WAVE_LDS];
  const int lane = threadIdx.x & 31;
  const int wave = threadIdx.x >> 5;
  const int tile = blockIdx.x * GEMV_WAVES + wave;
  if (tile * 16 >= M) return;                 // wave-uniform exit (EXEC stays all-1 elsewhere)

  float* wtile = lds + wave * GEMV_WAVE_LDS;  // 16 rows x 132 floats
  float* xs    = wtile + 16 * GEMV_ROWS;      // 128 floats of the x chunk
  const int r0 = tile * 16;

  // A-operand read base: lanes 0-15 -> row=lane, K offset 0/1;
  // lanes 16-31 -> row=lane-16, K offset 2/3 (ISA 16x4 fp32 A layout).
  const int koff = (lane >> 4) << 1;
  const float* arow = wtile + (lane & 15) * GEMV_ROWS + koff;
  const float* xoff = xs + koff;

  v8f acc = {};                               // 16x16 fp32 accumulator (8 VGPRs)

  for (int k0 = 0; k0 < K; k0 += GEMV_BK) {
    // stage x chunk: 32 lanes x float4 = 128 floats
    *(float4*)(xs + lane * 4) = *(const float4*)(v + k0 + lane * 4);
    // stage 16 rows of W, each row = one coalesced 512B burst across the wave
#pragma unroll
    for (int row = 0; row < 16; ++row) {
      int rg = r0 + row;
      if (rg >= M) rg = M - 1;                // clamp (duplicate last row on tail tile)
      *(float4*)(wtile + row * GEMV_ROWS + lane * 4) =
          *(const float4*)(W + (size_t)rg * (size_t)K + k0 + lane * 4);
    }
    // 32 WMMAs over this K chunk. Offsets (lane*132 + koff + kk) are even
    // floats -> 8B-aligned ds_load_b64, provably bank-conflict free.
#pragma unroll 8
    for (int kk = 0; kk < GEMV_BK; kk += 4) {
      v2f a = *(const v2f*)(arow + kk);       // W[row][kk+koff .. +1]
      v2f b = *(const v2f*)(xoff + kk);       // x[kk+koff .. +1] broadcast
      acc = __builtin_amdgcn_wmma_f32_16x16x4_f32(false, a, false, b,
                                                  (short)0, acc, false, false);
    }
  }

  // D layout: lane 0 holds rows r0..r0+7 in acc[0..7], lane 16 rows r0+8..r0+15.
  if (lane == 0 || lane == 16) {
    const int base = r0 + ((lane >> 4) << 3);
#pragma unroll
    for (int j = 0; j < 8; ++j) {
      int rg = base + j;
      if (rg < M) {
        float val = acc[j] + bias[rg];
        if (act == 1) val = fmaxf(val, 0.0f);
        y[rg] = val;
      }
    }
  }
}

// ---------------------------------------------------------------------------
// glue kernels
// ---------------------------------------------------------------------------
__global__ void embed_concat_kernel(const int* __restrict__ ids,
                                    const float* __restrict__ emb,
                                    const float* __restrict__ hidden,
                                    float* __restrict__ v1,
                                    float* __restrict__ v2) {
  int i = blockIdx.x * blockDim.x + threadIdx.x;       // 0..4095
  float e = emb[(size_t)ids[0] * H_SZ + i];
  v1[i] = e;                 // concat(embedded, hidden_flat) for attn
  v1[H_SZ + i] = hidden[i];
  v2[i] = e;                 // concat(embedded, attn_applied) front half
}

__device__ __forceinline__ float waveRedMax(float v) {
#pragma unroll
  for (int o = 16; o; o >>= 1) v = fmaxf(v, __shfl_xor(v, o, 32));
  return v;
}
__device__ __forceinline__ float waveRedSum(float v) {
#pragma unroll
  for (int o = 16; o; o >>= 1) v += __shfl_xor(v, o, 32);
  return v;
}
__device__ __forceinline__ float blockRed(float v, float* sm, bool isMax) {
  const int lane = threadIdx.x & 31, w = threadIdx.x >> 5;
  const int nw = (blockDim.x + 31) >> 5;
  v = isMax ? waveRedMax(v) : waveRedSum(v);
  if (lane == 0) sm[w] = v;
  __syncthreads();
  if (w == 0) {
    float t = (lane < nw) ? sm[lane] : (isMax ? -3.4e38f : 0.0f);
    t = isMax ? waveRedMax(t) : waveRedSum(t);
    if (lane == 0) sm[0] = t;
  }
  __syncthreads();
  float r = sm[0];
  __syncthreads();
  return r;
}

__global__ void softmax512_kernel(const float* __restrict__ logits,
                                  float* __restrict__ w_ws,
                                  float* __restrict__ w_out) {
  __shared__ float sm[32];
  const int t = threadIdx.x;                  // 512 threads
  float x = logits[t];
  float m = blockRed(x, sm, true);
  float e = __expf(x - m);
  float s = blockRed(e, sm, false);
  float w = e / s;
  w_ws[t] = w;
  w_out[t] = w;                               // attn_weights output
}

__global__ void attn_apply_kernel(const float* __restrict__ w,
                                  const float* __restrict__ latent,
                                  float* __restrict__ out) {
  int c = blockIdx.x * blockDim.x + threadIdx.x;       // 0..4095, coalesced over H
  float s = 0.0f;
  for (int l = 0; l < L_SZ; ++l)
    s = fmaf(w[l], latent[(size_t)l * H_SZ + c], s);
  out[c] = s;
}

__device__ __forceinline__ float sigmoidf(float x) { return 1.0f / (1.0f + __expf(-x)); }

__global__ void gru_combine_kernel(const float* __restrict__ gi_f,
                                   const float* __restrict__ gh_f,
                                   const float* __restrict__ gi_b,
                                   const float* __restrict__ gh_b,
                                   const float* __restrict__ hidden,
                                   float* __restrict__ newh_out,
                                   float* __restrict__ gru_out) {
  int t = blockIdx.x * blockDim.x + threadIdx.x;       // 0..4095
  int dir = t >> 11, j = t & (HH_SZ - 1);
  const float* gi = dir ? gi_b : gi_f;
  const float* gh = dir ? gh_b : gh_f;
  float h = hidden[dir * HH_SZ + j];
  float r = sigmoidf(gi[j] + gh[j]);
  float z = sigmoidf(gi[HH_SZ + j] + gh[HH_SZ + j]);
  float n = tanhf(gi[2 * HH_SZ + j] + r * gh[2 * HH_SZ + j]);
  float hn = (1.0f - z) * n + z * h;
  newh_out[t] = hn;                           // new_hidden output [h_f | h_b]
  gru_out[t] = hn;                            // concat(h_f, h_b) for out proj
}

__global__ void log_softmax_inplace_kernel(float* __restrict__ buf, int n) {
  __shared__ float sm[32];
  float m = -3.4e38f;
  for (int i = threadIdx.x; i < n; i += blockDim.x) m = fmaxf(m, buf[i]);
  m = blockRed(m, sm, true);
  float s = 0.0f;
  for (int i = threadIdx.x; i < n; i += blockDim.x) s += __expf(buf[i] - m);
  s = blockRed(s, sm, false);
  float lse = m + __logf(s);
  for (int i = threadIdx.x; i < n; i += blockDim.x) buf[i] -= lse;
}

// ---------------------------------------------------------------------------
static inline void gemv_launch(const float* W, const float* v, const float* b,
                               float* y, int M, int K, int act, hipStream_t s) {
  int tiles = (M + 15) / 16;
  int blocks = (tiles + GEMV_WAVES - 1) / GEMV_WAVES;
  gemv_wmma_kernel<<<blocks, GEMV_WAVES * 32, 0, s>>>(W, v, b, y, M, K, act);
}

extern "C" void kernel_launch(void* const* d_in, const int* in_sizes, int n_in,
                              void* d_out, int out_size, void* d_ws, size_t ws_size,
                              hipStream_t stream) {
  const int*   ids    = (const int*)d_in[0];
  const float* hidden = (const float*)d_in[1];   // [2,1,2048] -> flat 4096
  const float* latent = (const float*)d_in[2];   // [512,4096]
  const float* emb    = (const float*)d_in[3];   // [V,4096]
  const float* attn_W = (const float*)d_in[4];   // [512,8192]
  const float* attn_b = (const float*)d_in[5];
  const float* comb_W = (const float*)d_in[6];   // [4096,8192]
  const float* comb_b = (const float*)d_in[7];
  const float* Wih_f  = (const float*)d_in[8];   // [6144,4096]
  const float* Whh_f  = (const float*)d_in[9];   // [6144,2048]
  const float* bih_f  = (const float*)d_in[10];
  const float* bhh_f  = (const float*)d_in[11];
  const float* Wih_b  = (const float*)d_in[12];
  const float* Whh_b  = (const float*)d_in[13];
  const float* bih_b  = (const float*)d_in[14];
  const float* bhh_b  = (const float*)d_in[15];
  const float* out_W  = (const float*)d_in[16];  // [V,4096]
  const float* out_b  = (const float*)d_in[17];

  float* out = (float*)d_out;
  float* ws  = (float*)d_ws;

  // workspace layout (floats), all 16B-aligned offsets
  float* v1   = ws;            // 8192: [embedded | hidden_flat]
  float* v2   = ws + 8192;     // 8192: [embedded | attn_applied]
  float* alog = ws + 16384;    // 512 : attn logits
  float* aw   = ws + 16896;    // 512 : attn weights
  float* x    = ws + 17408;    // 4096: relu(comb)
  float* gi_f = ws + 21504;    // 6144
  float* gh_f = ws + 27648;    // 6144
  float* gi_b = ws + 33792;    // 6144
  float* gh_b = ws + 39936;    // 6144
  float* gout = ws + 46080;    // 4096: concat(h_f, h_b)

  float* logp  = out;                        // [V]   (log_softmax done in place)
  float* newh  = out + V_SZ;                 // [2,1,2048]
  float* awout = out + V_SZ + H_SZ;          // [1,512]

  embed_concat_kernel<<<16, 256, 0, stream>>>(ids, emb, hidden, v1, v2);
  gemv_launch(attn_W, v1, attn_b, alog, L_SZ, 2 * H_SZ, 0, stream);
  softmax512_kernel<<<1, 512, 0, stream>>>(alog, aw, awout);
  attn_apply_kernel<<<16, 256, 0, stream>>>(aw, latent, v2 + H_SZ);
  gemv_launch(comb_W, v2, comb_b, x, H_SZ, 2 * H_SZ, /*relu*/ 1, stream);
  gemv_launch(Wih_f, x, bih_f, gi_f, 3 * HH_SZ, H_SZ, 0, stream);
  gemv_launch(Whh_f, hidden, bhh_f, gh_f, 3 * HH_SZ, HH_SZ, 0, stream);
  gemv_launch(Wih_b, x, bih_b, gi_b, 3 * HH_SZ, H_SZ, 0, stream);
  gemv_launch(Whh_b, hidden + HH_SZ, bhh_b, gh_b, 3 * HH_SZ, HH_SZ, 0, stream);
  gru_combine_kernel<<<16, 256, 0, stream>>>(gi_f, gh_f, gi_b, gh_b, hidden, newh, gout);
  gemv_launch(out_W, gout, out_b, logp, V_SZ, H_SZ, 0, stream);
  log_softmax_inplace_kernel<<<1, 1024, 0, stream>>>(logp, V_SZ);
}